// OHEMLoss_42142219108551
// MI455X (gfx1250) — compile-verified
//
#include <hip/hip_runtime.h>
#include <cstdint>

// ---------------------------------------------------------------------------
// OHEM cross-entropy loss for gfx1250 (MI455X).
//   k_loss    : fused log-softmax NLL, 4 pixels/thread with b128 NT loads
//               -> loss[], n_valid, total fixed-point sum
//   k_hist x3 : radix histograms (11/11/9 bits of positive-float bit pattern);
//               loss tiles staged via global_load_async_to_lds_b128 (ASYNCcnt)
//   k_select x3: single-block descending scans -> exact k-th-largest bits
//   k_ohem    : masked fixed-point sum + count of loss >= thresh
//   k_final   : scalar division (n_valid==0 fallback = mean of all)
// All cross-block accumulation is integer -> deterministic under graph replay.
// ---------------------------------------------------------------------------

typedef float v4f __attribute__((ext_vector_type(4)));
typedef int   v4i __attribute__((ext_vector_type(4)));

constexpr int    kC  = 19;
constexpr int    kHW = 512 * 1024;          // 524288
constexpr int    kP  = 4 * kHW;             // 2097152 pixels
constexpr float  kFixScale = 67108864.0f;   // 2^26 fixed-point scale
constexpr double kInvFix   = 1.0 / 67108864.0;

// ctrl layout (uint32 view): [0]=n_valid [1]=kept_cnt [2]=sel prefix bits
// [3]=remaining rank [4]=thresh bits ; int64 view: [3]=total_sum [4]=kept_sum

// ---------------- kernel 1: per-pixel NLL + n_valid + total fixed-point sum
__global__ void k_loss(const float* __restrict__ logits,
                       const int* __restrict__ targets,
                       float* __restrict__ loss,
                       unsigned int* __restrict__ ctrl) {
  __shared__ unsigned long long sSum;
  __shared__ unsigned int sCnt;
  if (threadIdx.x == 0) { sSum = 0ull; sCnt = 0u; }
  __syncthreads();

  const int i  = blockIdx.x * blockDim.x + threadIdx.x;  // vec4 index; grid exact
  const int p0 = i << 2;
  const int n  = p0 >> 19;                               // p0 / kHW
  const int hw = p0 & (kHW - 1);
  const v4f* base = (const v4f*)(logits + (size_t)n * (size_t)(kC * kHW) + hw);

  v4f x[kC];
#pragma unroll
  for (int c = 0; c < kC; ++c)                            // 19 x b128, NT: stream,
    x[c] = __builtin_nontemporal_load(base + (size_t)c * (kHW / 4));  // keep L2 for loss

  const v4i t4 = __builtin_nontemporal_load((const v4i*)targets + i);

  v4f out;
  unsigned long long acc = 0ull;
  unsigned int cnt = 0u;
#pragma unroll
  for (int j = 0; j < 4; ++j) {
    float m = x[0][j];
#pragma unroll
    for (int c = 1; c < kC; ++c) m = fmaxf(m, x[c][j]);
    float s = 0.0f;
#pragma unroll
    for (int c = 0; c < kC; ++c) s += __expf(x[c][j] - m);   // v_exp_f32
    const int t = t4[j];
    float xt = x[0][j];
#pragma unroll
    for (int c = 1; c < kC; ++c) xt = (t == c) ? x[c][j] : xt;
    const float nll = (m - xt) + __logf(s);                  // v_log_f32
    out[j] = nll;
    acc += (unsigned long long)__float2ll_rn(nll * kFixScale);
    if (nll > 0.0f) ++cnt;
  }
  ((v4f*)loss)[i] = out;

  // wave32 pre-reduction, then one LDS atomic per wave
#pragma unroll
  for (int off = 16; off > 0; off >>= 1) {
    acc += __shfl_down(acc, off, 32);
    cnt += __shfl_down(cnt, off, 32);
  }
  if ((threadIdx.x & 31) == 0) {
    atomicAdd(&sSum, acc);
    atomicAdd(&sCnt, cnt);
  }
  __syncthreads();
  if (threadIdx.x == 0) {
    atomicAdd((unsigned long long*)(ctrl + 6), sSum);
    atomicAdd(ctrl + 0, sCnt);
  }
}

// ---------------- radix histogram pass (stages 0/1/2), LDS-privatized.
// Loss tiles are staged into LDS via CDNA5 async-to-LDS (ASYNCcnt) ops.
__global__ void k_hist(const float* __restrict__ loss,
                       const unsigned int* __restrict__ ctrl,
                       unsigned int* __restrict__ hist,
                       int stage) {
  __shared__ unsigned int h[2048];
  __shared__ alignas(16) float tile[1024];  // 256 threads * 4 floats
  for (int i = threadIdx.x; i < 2048; i += blockDim.x) h[i] = 0u;
  const unsigned int pref = (stage > 0) ? ctrl[2] : 0u;  // uniform
  __syncthreads();

  const unsigned int tid = threadIdx.x;
  // Low 32 bits of the generic pointer to an LDS object == wave-relative LDS
  // byte address (flat-aperture layout) == what VDST of the async op takes.
  const unsigned int ldsOff =
      (unsigned int)(uintptr_t)(&tile[0]) + tid * 16u;

  const long long step = (long long)gridDim.x * 1024;   // 256 thr * 4 floats
  for (long long base = (long long)blockIdx.x * 1024; base < kP; base += step) {
    const unsigned long long gaddr =
        (unsigned long long)(uintptr_t)(loss + base + tid * 4);
    asm volatile("global_load_async_to_lds_b128 %0, %1, off"
                 :: "v"(ldsOff), "v"(gaddr) : "memory");
    asm volatile("s_wait_asynccnt 0" ::: "memory");
    __syncthreads();
#pragma unroll
    for (int j = 0; j < 4; ++j) {
      const float v = tile[tid * 4 + j];
      if (v > 0.0f) {
        const unsigned int bits = __float_as_uint(v);
        int bin = -1;
        if (stage == 0)      bin = (int)(bits >> 20);                       // bits[30:20]
        else if (stage == 1) { if ((bits >> 20) == pref) bin = (int)((bits >> 9) & 0x7FFu); }
        else                 { if ((bits >> 9)  == pref) bin = (int)(bits & 0x1FFu); }
        if (bin >= 0) atomicAdd(&h[bin], 1u);
      }
    }
    __syncthreads();
  }
  for (int i = threadIdx.x; i < 2048; i += blockDim.x) {
    const unsigned int c = h[i];
    if (c) atomicAdd(&hist[i], c);
  }
}

// ---------------- single-block descending (suffix) scan + rank selection
__device__ __forceinline__ void write_sel(unsigned int* ctrl, int stage,
                                          unsigned int bin, unsigned int rank) {
  if (stage == 0)      { ctrl[2] = bin;                    ctrl[3] = rank; }
  else if (stage == 1) { ctrl[2] = (ctrl[2] << 11) | bin;  ctrl[3] = rank; }
  else                 { ctrl[4] = (ctrl[2] << 9)  | bin; }
}

__global__ void k_select(const unsigned int* __restrict__ hist, int nbins,
                         int stage, unsigned int* __restrict__ ctrl) {
  __shared__ unsigned int s[2048];
  const unsigned int nv = ctrl[0];
  if (nv == 0u) {  // no valid pixels: empty mask; k_final uses fallback mean
    if (stage == 2 && threadIdx.x == 0) ctrl[4] = 0x7F800000u;  // +inf
    return;
  }
  unsigned int K;
  if (stage == 0) {  // n_keep with the reference's f32 truncation semantics
    K = (unsigned int)(0.7f * (float)nv);
    if (K < 100000u) K = 100000u;
    if (K > nv) K = nv;
  } else {
    K = ctrl[3];
  }
  const int i0 = threadIdx.x, i1 = threadIdx.x + 1024;
  s[i0] = (i0 < nbins) ? hist[nbins - 1 - i0] : 0u;   // reversed: s[0] = top bin
  s[i1] = (i1 < nbins) ? hist[nbins - 1 - i1] : 0u;
  __syncthreads();
  for (int off = 1; off < 2048; off <<= 1) {          // Hillis-Steele inclusive scan
    const unsigned int v0 = (i0 >= off) ? s[i0 - off] : 0u;
    const unsigned int v1 = s[i1 - off];              // i1 >= 1024 >= off
    __syncthreads();
    s[i0] += v0; s[i1] += v1;
    __syncthreads();
  }
  const unsigned int p0 = (i0 > 0) ? s[i0 - 1] : 0u;
  const unsigned int p1 = s[i1 - 1];
  if (s[i0] >= K && p0 < K) write_sel(ctrl, stage, (unsigned int)(nbins - 1 - i0), K - p0);
  if (s[i1] >= K && p1 < K) write_sel(ctrl, stage, (unsigned int)(nbins - 1 - i1), K - p1);
}

// ---------------- masked fixed-point sum of kept losses (b128 reads)
__global__ void k_ohem(const float* __restrict__ loss,
                       unsigned int* __restrict__ ctrl) {
  __shared__ unsigned long long sSum;
  __shared__ unsigned int sCnt;
  if (threadIdx.x == 0) { sSum = 0ull; sCnt = 0u; }
  __syncthreads();
  const float th = __uint_as_float(ctrl[4]);
  long long acc = 0;
  unsigned int cnt = 0;
  const int stride = gridDim.x * blockDim.x;
  const v4f* lv = (const v4f*)loss;
  for (int i = blockIdx.x * blockDim.x + threadIdx.x; i < kP / 4; i += stride) {
    const v4f v = lv[i];
#pragma unroll
    for (int j = 0; j < 4; ++j) {
      if (v[j] >= th) { acc += __float2ll_rn(v[j] * kFixScale); ++cnt; }
    }
  }
  unsigned long long accU = (unsigned long long)acc;
#pragma unroll
  for (int off = 16; off > 0; off >>= 1) {
    accU += __shfl_down(accU, off, 32);
    cnt  += __shfl_down(cnt, off, 32);
  }
  if ((threadIdx.x & 31) == 0) {
    atomicAdd(&sSum, accU);
    atomicAdd(&sCnt, cnt);
  }
  __syncthreads();
  if (threadIdx.x == 0) {
    atomicAdd((unsigned long long*)(ctrl + 8), sSum);
    atomicAdd(ctrl + 1, sCnt);
  }
}

// ---------------- finalize scalar
__global__ void k_final(const unsigned int* __restrict__ ctrl,
                        float* __restrict__ out) {
  const long long total = ((const long long*)ctrl)[3];
  const long long kept  = ((const long long*)ctrl)[4];
  const unsigned int nv = ctrl[0];
  unsigned int kc = ctrl[1];
  double r;
  if (nv > 0u) {
    if (kc < 1u) kc = 1u;
    r = ((double)kept * kInvFix) / (double)kc;
  } else {
    r = ((double)total * kInvFix) / (double)kP;
  }
  out[0] = (float)r;
}

extern "C" void kernel_launch(void* const* d_in, const int* in_sizes, int n_in,
                              void* d_out, int out_size, void* d_ws, size_t ws_size,
                              hipStream_t stream) {
  (void)in_sizes; (void)n_in; (void)out_size; (void)ws_size;
  const float* logits  = (const float*)d_in[0];
  const int*   targets = (const int*)d_in[1];
  float* out = (float*)d_out;

  // workspace layout (4-byte units): loss[kP], hA[2048], hB[2048], hC[2048], ctrl[32]
  float* loss = (float*)d_ws;
  unsigned int* hA   = (unsigned int*)d_ws + kP;
  unsigned int* hB   = hA + 2048;
  unsigned int* hC   = hB + 2048;
  unsigned int* ctrl = hC + 2048;

  hipMemsetAsync(hA, 0, (size_t)(3 * 2048 + 32) * sizeof(unsigned int), stream);

  k_loss  <<<kP / 1024, 256, 0, stream>>>(logits, targets, loss, ctrl);
  k_hist  <<<512, 256, 0, stream>>>(loss, ctrl, hA, 0);
  k_select<<<1, 1024, 0, stream>>>(hA, 2048, 0, ctrl);
  k_hist  <<<512, 256, 0, stream>>>(loss, ctrl, hB, 1);
  k_select<<<1, 1024, 0, stream>>>(hB, 2048, 1, ctrl);
  k_hist  <<<512, 256, 0, stream>>>(loss, ctrl, hC, 2);
  k_select<<<1, 1024, 0, stream>>>(hC, 512, 2, ctrl);
  k_ohem  <<<512, 256, 0, stream>>>(loss, ctrl);
  k_final <<<1, 1, 0, stream>>>(ctrl, out);
}